// TransformationsModule_59906203845005
// MI455X (gfx1250) — compile-verified
//
#include <hip/hip_runtime.h>
#include <hip/hip_bf16.h>
#include <stdint.h>

// ---------------- problem constants (fixed by the reference) ----------------
#define T_LEN    32768
#define LCH      8          // input channels z0..z7
#define TILE     1024       // t-positions per block
#define HALO     3          // max conv half-width (gaussian w=7)
#define NTHREADS 256        // 8 wave32s
#define SEGS     (T_LEN / TILE)
#define EPSF     1e-6f

// ---------------- CDNA5 TDM availability (device pass only) ----------------
#if defined(__has_builtin)
#  if __has_builtin(__builtin_amdgcn_tensor_load_to_lds) && __has_builtin(__builtin_amdgcn_s_wait_tensorcnt)
#    define USE_TDM 1
#  endif
#endif
#ifndef USE_TDM
#  define USE_TDM 0
#endif

#if USE_TDM
typedef unsigned int v4u __attribute__((ext_vector_type(4)));
typedef int          v8i __attribute__((ext_vector_type(8)));
typedef int          v4i __attribute__((ext_vector_type(4)));
#endif

// ---------------- fallback: per-lane async global->LDS (verified asm) -------
__device__ __forceinline__ void async_load_f4(const float* g, float* lds) {
  unsigned lds_off = (unsigned)(uintptr_t)lds;  // low 32 bits of generic LDS ptr = LDS offset
  asm volatile("global_load_async_to_lds_b128 %0, %1, off"
               :: "v"(lds_off), "v"(g)
               : "memory");
}
__device__ __forceinline__ void wait_asynccnt0() {
  asm volatile("s_wait_asynccnt 0x0" ::: "memory");
}

// ---------------- protected math, faithful to the reference ----------------
__device__ __forceinline__ void compute_s(const float* zp, float& s0, float& s1) {
  float z0 = zp[0], z1 = zp[1], z2 = zp[2], z3 = zp[3], z4 = zp[4];
  s0 = sinf(z0 * z1) + z2;
  float th  = tanhf(z4);
  float sgn = (th > 0.f) ? 1.f : ((th < 0.f) ? -1.f : 0.f);   // jnp.sign (sign(0)=0)
  float den = (fabsf(th) > EPSF) ? th : fmaf(sgn, EPSF, EPSF); // th<0 -> exactly 0 (matches ref)
  s1 = z3 / den;
}

// ============================ Kernel A ============================
__global__ __launch_bounds__(NTHREADS) void k_compute(
    const float* __restrict__ in, float* __restrict__ out,
    float* __restrict__ partials) {
  __shared__ __align__(16) float sh_in[(TILE + 2 * HALO) * LCH];  // 33 KB
  __shared__ float sh_s0[TILE + 2 * HALO];
  __shared__ float sh_s1[TILE + 2 * HALO];
  __shared__ float red[8][8];

  const int tid = threadIdx.x;
  const int seg = blockIdx.x;
  const int b   = blockIdx.y;
  const int t0  = seg * TILE;
  const float* row = in + (size_t)b * T_LEN * LCH;

#if USE_TDM
  // ---- TDM: one descriptor stages the whole (TILE+6) x 8 f32 tile.
  // tensor_dim0 = elements to end of row -> HW zero-fills reads past it
  // (right-edge halo). Left edge (seg 0) zero-filled with ds stores.
  {
    int start_t = t0 - HALO;
    int skip_t  = 0;
    if (start_t < 0) { skip_t = -start_t; start_t = 0; }
    if (tid < 32) {  // one TDM issue (wave 0); EXEC is ignored by tensor ops
      unsigned lds_base = (unsigned)(uintptr_t)&sh_in[0] + (unsigned)(skip_t * LCH * 4);
      unsigned long long gaddr =
          (unsigned long long)(uintptr_t)(row + (size_t)start_t * LCH);
      unsigned td0   = (unsigned)(T_LEN - start_t) * LCH;            // tensor_dim0 (elems)
      unsigned tile0 = (unsigned)((TILE + 2 * HALO - skip_t) * LCH); // tile_dim0 <= 8240
      v4u g0; v8i g1; v4i gz = {0, 0, 0, 0};
      g0.x = 1u;                                        // count=1 (valid user D#)
      g0.y = lds_base;                                  // lds_addr [63:32]
      g0.z = (unsigned)gaddr;                           // global_addr[31:0]
      g0.w = (unsigned)((gaddr >> 32) & 0x01FFFFFFu)    // global_addr[56:32]
             | 0x80000000u;                             // type=2 ("image") [127:126]
      g1[0] = (int)(2u << 16);                          // data_size=2 -> 4 bytes
      g1[1] = (int)((td0 & 0xFFFFu) << 16);             // tensor_dim0[15:0]
      g1[2] = (int)(((td0 >> 16) & 0xFFFFu)             // tensor_dim0[31:16]
                    | (1u << 16));                      // tensor_dim1=1 (unused, 1D tile)
      g1[3] = (int)(tile0 << 16);                       // tile_dim0
      g1[4] = 0;                                        // tile_dim1=0, tile_dim2=0 (1D)
      g1[5] = (int)td0;                                 // tensor_dim0_stride (unused for 1D)
      g1[6] = 0;
      g1[7] = 0;
#if defined(__clang_major__) && (__clang_major__ >= 23)
      v8i g4 = {0, 0, 0, 0, 0, 0, 0, 0};
      __builtin_amdgcn_tensor_load_to_lds(g0, g1, gz, gz, g4, 0);
#else
      __builtin_amdgcn_tensor_load_to_lds(g0, g1, gz, gz, 0);
#endif
      __builtin_amdgcn_s_wait_tensorcnt(0);
    }
    if (skip_t && tid < skip_t * 2) {  // zero left-halo float4 slots (seg 0 only)
      *(float4*)&sh_in[tid * 4] = make_float4(0.f, 0.f, 0.f, 0.f);
    }
  }
  __syncthreads();
#else
  // ---- fallback: per-lane async loads, OOB halo zero-filled
  const int NCHUNK = (TILE + 2 * HALO) * 2;
  for (int c = tid; c < NCHUNK; c += NTHREADS) {
    int t = t0 - HALO + (c >> 1);
    if (t >= 0 && t < T_LEN) {
      const float* g = row + (size_t)t * LCH + (c & 1) * 4;
      async_load_f4(g, &sh_in[c * 4]);
    }
  }
  wait_asynccnt0();
  for (int c = tid; c < NCHUNK; c += NTHREADS) {
    int t = t0 - HALO + (c >> 1);
    if (t < 0 || t >= T_LEN) *(float4*)&sh_in[c * 4] = make_float4(0.f, 0.f, 0.f, 0.f);
  }
  __syncthreads();
#endif

  // Gaussian kernel w=7, std=1.5 : exp(-d^2/4.5)/sum (computed like the ref)
  float gk[7];
  {
    float s = 0.f;
    #pragma unroll
    for (int i = 0; i < 7; ++i) { float d = (float)(i - 3); gk[i] = expf(-d * d / 4.5f); s += gk[i]; }
    #pragma unroll
    for (int i = 0; i < 7; ++i) gk[i] /= s;
  }

  // ---- pass 1: s0/s1 into LDS; c2/c3 direct
  float sum2 = 0.f, sq2 = 0.f, sum3 = 0.f, sq3 = 0.f;
  float c2r[TILE / NTHREADS], c3r[TILE / NTHREADS];
  #pragma unroll
  for (int k = 0; k < TILE / NTHREADS; ++k) {
    int idx = k * NTHREADS + tid;
    int w   = idx + HALO;
    const float* zp = &sh_in[w * LCH];
    float s0v, s1v;
    compute_s(zp, s0v, s1v);
    sh_s0[w] = s0v;
    sh_s1[w] = s1v;
    float z5 = zp[5], z6 = zp[6], z7 = zp[7];
    float c2 = logf(fabsf(z5) + EPSF) * expf(fminf(fmaxf(z6, -40.f), -20.f));
    float c3 = z7 * z7;
    c2r[k] = c2; c3r[k] = c3;
    sum2 += c2; sq2 += c2 * c2;
    sum3 += c3; sq3 += c3 * c3;
  }
  if (tid < 2 * HALO) {  // halo s0/s1 (zero outside [0,T): conv zero-padding)
    int w = (tid < HALO) ? tid : (TILE + HALO + (tid - HALO));
    const float* zp = &sh_in[w * LCH];
    float s0v, s1v;
    compute_s(zp, s0v, s1v);
    int t = t0 - HALO + w;
    bool ok = (t >= 0) && (t < T_LEN);
    sh_s0[w] = ok ? s0v : 0.f;
    sh_s1[w] = ok ? s1v : 0.f;
  }
  __syncthreads();

  // ---- pass 2: convs from LDS, write un-normalized float4 output
  float sum0 = 0.f, sq0 = 0.f, sum1 = 0.f, sq1 = 0.f;
  float4* orow = (float4*)out + (size_t)b * T_LEN + t0;
  #pragma unroll
  for (int k = 0; k < TILE / NTHREADS; ++k) {
    int idx = k * NTHREADS + tid;
    float a = 0.f;
    #pragma unroll
    for (int i = 0; i < 5; ++i) a += sh_s0[idx + 1 + i];   // t-2..t+2
    float c0 = 0.2f * a;
    float c1 = 0.f;
    #pragma unroll
    for (int i = 0; i < 7; ++i) c1 = fmaf(sh_s1[idx + i], gk[i], c1);  // t-3..t+3
    sum0 += c0; sq0 += c0 * c0;
    sum1 += c1; sq1 += c1 * c1;
    orow[idx] = make_float4(c0, c1, c2r[k], c3r[k]);
  }

  // ---- block reduction: wave32 shfl_xor, then cross-wave via LDS
  float vals[8] = {sum0, sq0, sum1, sq1, sum2, sq2, sum3, sq3};
  #pragma unroll
  for (int i = 0; i < 8; ++i) {
    float v = vals[i];
    #pragma unroll
    for (int off = 16; off > 0; off >>= 1) v += __shfl_xor(v, off, 32);
    vals[i] = v;
  }
  int lane = tid & 31, wv = tid >> 5;
  if (lane == 0) {
    #pragma unroll
    for (int i = 0; i < 8; ++i) red[wv][i] = vals[i];
  }
  __syncthreads();
  if (tid < 8) {
    float s = 0.f;
    #pragma unroll
    for (int w = 0; w < 8; ++w) s += red[w][tid];
    partials[((size_t)b * SEGS + seg) * 8 + tid] = s;
  }
}

// ============================ Kernel B ============================
__global__ void k_stats(const float* __restrict__ partials, float* __restrict__ stats) {
  const int b = blockIdx.x;
  __shared__ float tot[8];
  int tid = threadIdx.x;
  if (tid < 8) {
    float s = 0.f;
    const float* p = partials + (size_t)b * SEGS * 8 + tid;
    for (int seg = 0; seg < SEGS; ++seg) s += p[seg * 8];
    tot[tid] = s;
  }
  __syncthreads();
  if (tid < 4) {
    const float SC[4] = {1.0f, 0.5f, 2.0f, 1.5f};
    float mean = tot[tid * 2] / (float)T_LEN;
    float var  = (tot[tid * 2 + 1] - (float)T_LEN * mean * mean) / (float)(T_LEN - 1);
    var = fmaxf(var, 0.f);
    stats[b * 8 + tid * 2]     = mean;
    stats[b * 8 + tid * 2 + 1] = SC[tid] / (sqrtf(var) + EPSF);
  }
}

// ============================ Kernel C ============================
__global__ __launch_bounds__(256) void k_norm(float* __restrict__ out,
                                              const float* __restrict__ stats) {
  size_t i = (size_t)blockIdx.x * 256 + threadIdx.x;   // (b,t) index
  int b = (int)(i >> 15);                              // T = 2^15
  const float* st = stats + b * 8;
  float4 v = ((float4*)out)[i];
  v.x = (v.x - st[0]) * st[1];
  v.y = (v.y - st[2]) * st[3];
  v.z = (v.z - st[4]) * st[5];
  v.w = (v.w - st[6]) * st[7];
  ((float4*)out)[i] = v;
}

// ============================ launch ============================
extern "C" void kernel_launch(void* const* d_in, const int* in_sizes, int n_in,
                              void* d_out, int out_size, void* d_ws, size_t ws_size,
                              hipStream_t stream) {
  const float* in = (const float*)d_in[0];
  float* out = (float*)d_out;
  const int B = in_sizes[0] / (T_LEN * LCH);   // 128

  float* partials = (float*)d_ws;                       // B*SEGS*8 floats (128 KB)
  float* stats    = partials + (size_t)B * SEGS * 8;    // B*8 floats

  dim3 gridA(SEGS, B);
  k_compute<<<gridA, NTHREADS, 0, stream>>>(in, out, partials);
  k_stats<<<B, 32, 0, stream>>>(partials, stats);
  size_t nBT = (size_t)B * T_LEN;
  k_norm<<<(unsigned)(nBT / 256), 256, 0, stream>>>(out, stats);
}